// JosephProjector3D_24215025615244
// MI455X (gfx1250) — compile-verified
//
#include <hip/hip_runtime.h>
#include <math.h>

#define NW 128   // W (x)
#define NH 128   // H (y)
#define ND 128   // D (z), contiguous axis of vol
#define NA 96    // angles
#define NU 128   // detector u
#define NV 128   // detector v
#define NSTEPS 128

#define LDS_STRIDE 132   // 128 z + 4 pad floats (bank de-confliction)

typedef __attribute__((ext_vector_type(2))) float v2f;
typedef __attribute__((ext_vector_type(8))) float v8f;

__global__ __launch_bounds__(256) void joseph_fused_kernel(
    const float* __restrict__ vol,     // (W,H,D), z contiguous
    const float* __restrict__ angles,  // (A)
    float* __restrict__ out,           // (U,A,V)
    float Tlen, float dt)
{
    __shared__ float Plds[16 * LDS_STRIDE];   // P[u_local][z]

    const int a     = blockIdx.y;
    const int utile = blockIdx.x;
    const int lane  = threadIdx.x & 31;       // wave32
    const int wave  = threadIdx.x >> 5;       // 0..7

    const float ang = angles[a];
    const float ct = cosf(ang);
    const float st = sinf(ang);

    // ---------------- Phase 1: ray-march partial sums P[z][u] ----------------
    // One wave per u (2 u's per wave); lane covers z = 4*lane .. 4*lane+3.
    for (int uu = 0; uu < 2; ++uu) {
        const int ulocal = wave * 2 + uu;              // 0..15
        const int u      = utile * 16 + ulocal;
        const float uc   = (float)u - 0.5f * (float)(NU - 1);
        // x_idx = -u*sin + t*cos + (W-1)/2 ; y_idx = u*cos + t*sin + (H-1)/2
        const float xb = fmaf(-uc, st, 0.5f * (float)(NW - 1));
        const float yb = fmaf( uc, ct, 0.5f * (float)(NH - 1));
        const int zb = lane * 4;

        float4 acc = make_float4(0.f, 0.f, 0.f, 0.f);

        #pragma unroll 2
        for (int i = 0; i < NSTEPS; ++i) {
            const float t  = fmaf((float)i + 0.5f, dt, -Tlen);
            const float xi = fmaf(t, ct, xb);
            const float yi = fmaf(t, st, yb);

            const float xf = floorf(xi), yf = floorf(yi);
            const float fx = xi - xf,    fy = yi - yf;
            const int x0 = (int)xf, y0 = (int)yf;
            const int x1 = x0 + 1,  y1 = y0 + 1;

            const float wx0 = (x0 >= 0 && x0 < NW) ? (1.0f - fx) : 0.0f;
            const float wx1 = (x1 >= 0 && x1 < NW) ? fx          : 0.0f;
            const float wy0 = (y0 >= 0 && y0 < NH) ? (1.0f - fy) : 0.0f;
            const float wy1 = (y1 >= 0 && y1 < NH) ? fy          : 0.0f;

            const int cx0 = min(max(x0, 0), NW - 1);
            const int cx1 = min(max(x1, 0), NW - 1);
            const int cy0 = min(max(y0, 0), NH - 1);
            const int cy1 = min(max(y1, 0), NH - 1);

            // 4 contiguous z-lines (global_load_b128 each, L2-resident)
            const float4 L00 = *(const float4*)(vol + ((cx0 * NH + cy0) * ND + zb));
            const float4 L10 = *(const float4*)(vol + ((cx1 * NH + cy0) * ND + zb));
            const float4 L01 = *(const float4*)(vol + ((cx0 * NH + cy1) * ND + zb));
            const float4 L11 = *(const float4*)(vol + ((cx1 * NH + cy1) * ND + zb));

            const float w00 = wy0 * wx0;   // pairs (x0,y0)
            const float w10 = wy0 * wx1;   // (x1,y0)
            const float w01 = wy1 * wx0;   // (x0,y1)
            const float w11 = wy1 * wx1;   // (x1,y1)

            acc.x = fmaf(w00, L00.x, acc.x); acc.y = fmaf(w00, L00.y, acc.y);
            acc.z = fmaf(w00, L00.z, acc.z); acc.w = fmaf(w00, L00.w, acc.w);
            acc.x = fmaf(w10, L10.x, acc.x); acc.y = fmaf(w10, L10.y, acc.y);
            acc.z = fmaf(w10, L10.z, acc.z); acc.w = fmaf(w10, L10.w, acc.w);
            acc.x = fmaf(w01, L01.x, acc.x); acc.y = fmaf(w01, L01.y, acc.y);
            acc.z = fmaf(w01, L01.z, acc.z); acc.w = fmaf(w01, L01.w, acc.w);
            acc.x = fmaf(w11, L11.x, acc.x); acc.y = fmaf(w11, L11.y, acc.y);
            acc.z = fmaf(w11, L11.z, acc.z); acc.w = fmaf(w11, L11.w, acc.w);
        }
        *(float4*)&Plds[ulocal * LDS_STRIDE + zb] = acc;
    }
    __syncthreads();

    // ---------------- Phase 2: z-interp GEMM  out[v,u] = dt * Z[v,:] @ P[:,u] ----------------
    // One wave per 16-v tile. v_wmma_f32_16x16x4_f32, K swept 4 at a time (32 MACs).
    const int vtile = wave;
    const int nidx  = lane & 15;          // A: row M ; B: col N ; D: col N
    const int kofs  = (lane >> 4) * 2;    // K sub-offset per lane half (ISA 16x4 layout)

    // Z-row (wz0 @ z0, wz1 @ z1) for this lane's output row v  (general SD/SV form)
    const int v = vtile * 16 + nidx;
    const float vdet = ((float)v - 0.5f * (float)(NV - 1));          // * SV (=1)
    const float zidx = vdet /* / SD */ + 0.5f * (float)(ND - 1);
    const float zfl  = floorf(zidx);
    const float zfr  = zidx - zfl;
    const int z0i = (int)zfl, z1i = z0i + 1;
    const float wz0 = (z0i >= 0 && z0i < ND) ? (1.0f - zfr) : 0.0f;
    const float wz1 = (z1i >= 0 && z1i < ND) ? zfr          : 0.0f;
    const int z0c = min(max(z0i, 0), ND - 1);
    const int z1c = min(max(z1i, 0), ND - 1);

    const float* brow = &Plds[nidx * LDS_STRIDE];   // P[u=nidx][z], conflict-free reads

    v8f c = {};
    for (int kb = 0; kb < ND / 4; ++kb) {
        v2f afrag, bfrag;
        #pragma unroll
        for (int j = 0; j < 2; ++j) {
            const int zk = kb * 4 + kofs + j;
            float av = (zk == z0c) ? wz0 : 0.0f;
            av      += (zk == z1c) ? wz1 : 0.0f;
            afrag[j] = av;
            bfrag[j] = brow[zk];
        }
        // (neg_a, A, neg_b, B, c_mod, C, reuse_a, reuse_b)
        c = __builtin_amdgcn_wmma_f32_16x16x4_f32(
                false, afrag, false, bfrag, (short)0, c, false, false);
    }

    // D layout: VGPR i -> rows M=i (lanes 0-15) / M=i+8 (lanes 16-31), N = lane&15.
    const int u     = utile * 16 + nidx;
    const int vbase = vtile * 16 + (lane >> 4) * 8;
    float* dst = out + ((size_t)u * NA + a) * NV + vbase;   // out is (U, A, V)
    float4 lo = make_float4(c[0] * dt, c[1] * dt, c[2] * dt, c[3] * dt);
    float4 hi = make_float4(c[4] * dt, c[5] * dt, c[6] * dt, c[7] * dt);
    *(float4*)(dst)     = lo;
    *(float4*)(dst + 4) = hi;
}

extern "C" void kernel_launch(void* const* d_in, const int* in_sizes, int n_in,
                              void* d_out, int out_size, void* d_ws, size_t ws_size,
                              hipStream_t stream) {
    const float* vol    = (const float*)d_in[0];   // (1,1,W,H,D) f32
    const float* angles = (const float*)d_in[1];   // (A) f32
    float* out = (float*)d_out;                    // (1,1,U,A,V) f32

    // T = 0.5*sqrt(((W-1)SX)^2 + ((H-1)SY)^2), delta_t = 2T/N  (match f64->f32 cast of reference)
    const double Td = 0.5 * sqrt((double)(NW - 1) * (NW - 1) + (double)(NH - 1) * (NH - 1));
    const float Tlen = (float)Td;
    const float dt   = (float)(2.0 * Td / (double)NSTEPS);

    dim3 grid(NU / 16, NA);   // (8 u-tiles, 96 angles)
    joseph_fused_kernel<<<grid, 256, 0, stream>>>(vol, angles, out, Tlen, dt);
}